// L3WithDlossIoULoss_8143257993253
// MI455X (gfx1250) — compile-verified
//
#include <hip/hip_runtime.h>

typedef float v2f __attribute__((ext_vector_type(2)));
typedef float v4f __attribute__((ext_vector_type(4)));
typedef float v8f __attribute__((ext_vector_type(8)));

#define HH 512
#define WW 512
#define HWPX (HH * WW)           // 262144 pixels per image
#define BB 32                    // batch
#define NCHUNK 64                // chunks per image
#define CHUNK_PX (HWPX / NCHUNK) // 4096 px per block
#define NTHREADS 256
#define NQ 16                    // quantity columns in partial rows

// Partial row layout (per block / per chunk):
//  [0]=sum_p [1]=sum_t [2]=sum_pt [3]=sum_xp [4]=sum_yp [5]=sum_xt [6]=sum_yt [7]=0
//  [8]=pminx [9]=pminy [10]=pmaxx [11]=pmaxy [12]=tminx [13]=tminy [14]=tmaxx [15]=tmaxy

__device__ __forceinline__ float sigmoid_fast(float z) {
  // 1/(1+exp(-z)) with raw v_exp_f32 + v_rcp_f32 (1-ulp), no NR refinement
  return __builtin_amdgcn_rcpf(1.0f + __expf(-z));
}

__global__ __launch_bounds__(NTHREADS) void
stage1(const float* __restrict__ pred_log, const float* __restrict__ target,
       float* __restrict__ partials) {
  __shared__ float P[NTHREADS][NQ];  // per-thread sum partials (matrix for WMMA reduce)
  __shared__ float MMX[8][8];        // per-wave min/max partials

  const int tid = threadIdx.x;
  const int chunk = blockIdx.x;
  const int img = blockIdx.y;
  const long base = (long)img * HWPX + (long)chunk * CHUNK_PX;
  const v4f* __restrict__ p4 = (const v4f*)(pred_log + base);
  const v4f* __restrict__ t4 = (const v4f*)(target + base);

  // Hoist all streaming loads: 8 x b128 in flight per thread (single-use -> NT hint)
  v4f pv[4], tv[4];
#pragma unroll
  for (int it = 0; it < 4; ++it) pv[it] = __builtin_nontemporal_load(&p4[it * NTHREADS + tid]);
#pragma unroll
  for (int it = 0; it < 4; ++it) tv[it] = __builtin_nontemporal_load(&t4[it * NTHREADS + tid]);

  float s_p = 0.f, s_t = 0.f, s_pt = 0.f;
  float s_xp = 0.f, s_yp = 0.f, s_xt = 0.f, s_yt = 0.f;
  float pminx = INFINITY, pminy = INFINITY, pmaxx = -INFINITY, pmaxy = -INFINITY;
  float tminx = INFINITY, tminy = INFINITY, tmaxx = -INFINITY, tmaxy = -INFINITY;
  const float inv = 1.0f / 512.0f;
  const float tau = 0.01f;

#pragma unroll
  for (int it = 0; it < 4; ++it) {
    const int vec = it * NTHREADS + tid;
    const int pix = chunk * CHUNK_PX + vec * 4;  // quad start; 4 px share one row
    const float y = (float)(pix >> 9) * inv;
    const float x0 = (float)(pix & 511) * inv;
    const float x1 = x0 + inv, x2 = x0 + 2.0f * inv, x3 = x0 + 3.0f * inv;

    const float p0 = sigmoid_fast(pv[it][0]);
    const float p1 = sigmoid_fast(pv[it][1]);
    const float p2 = sigmoid_fast(pv[it][2]);
    const float p3 = sigmoid_fast(pv[it][3]);
    const float t0 = tv[it][0], t1 = tv[it][1], t2 = tv[it][2], t3 = tv[it][3];

    // quad sums (reused for the y-moment)
    const float sp4 = (p0 + p1) + (p2 + p3);
    const float st4 = (t0 + t1) + (t2 + t3);
    s_p += sp4;
    s_t += st4;
    s_yp = fmaf(y, sp4, s_yp);
    s_yt = fmaf(y, st4, s_yt);
    s_pt = fmaf(p0, t0, fmaf(p1, t1, fmaf(p2, t2, fmaf(p3, t3, s_pt))));
    s_xp = fmaf(x0, p0, fmaf(x1, p1, fmaf(x2, p2, fmaf(x3, p3, s_xp))));
    s_xt = fmaf(x0, t0, fmaf(x1, t1, fmaf(x2, t2, fmaf(x3, t3, s_xt))));

    // pred mask bbox: first/last-active select chains + row any-active
    {
      const bool a0 = p0 > tau, a1 = p1 > tau, a2 = p2 > tau, a3 = p3 > tau;
      const float mnc = a0 ? x0 : (a1 ? x1 : (a2 ? x2 : (a3 ? x3 : INFINITY)));
      const float mxc = a3 ? x3 : (a2 ? x2 : (a1 ? x1 : (a0 ? x0 : -INFINITY)));
      const bool any = a0 | a1 | a2 | a3;
      pminx = fminf(pminx, mnc);
      pmaxx = fmaxf(pmaxx, mxc);
      pminy = fminf(pminy, any ? y : INFINITY);
      pmaxy = fmaxf(pmaxy, any ? y : -INFINITY);
    }
    // target mask bbox
    {
      const bool a0 = t0 > tau, a1 = t1 > tau, a2 = t2 > tau, a3 = t3 > tau;
      const float mnc = a0 ? x0 : (a1 ? x1 : (a2 ? x2 : (a3 ? x3 : INFINITY)));
      const float mxc = a3 ? x3 : (a2 ? x2 : (a1 ? x1 : (a0 ? x0 : -INFINITY)));
      const bool any = a0 | a1 | a2 | a3;
      tminx = fminf(tminx, mnc);
      tmaxx = fmaxf(tmaxx, mxc);
      tminy = fminf(tminy, any ? y : INFINITY);
      tmaxy = fmaxf(tmaxy, any ? y : -INFINITY);
    }
  }

  // --- stash per-thread sum partials into the LDS matrix for the WMMA reduce ---
  P[tid][0] = s_p;  P[tid][1] = s_t;  P[tid][2] = s_pt; P[tid][3] = s_xp;
  P[tid][4] = s_yp; P[tid][5] = s_xt; P[tid][6] = s_yt;
#pragma unroll
  for (int q = 7; q < NQ; ++q) P[tid][q] = 0.0f;

  // --- wave-level min/max reduce via shuffles ---
  const int lane = tid & 31;
  const int wave = tid >> 5;
#pragma unroll
  for (int off = 16; off > 0; off >>= 1) {
    pminx = fminf(pminx, __shfl_xor(pminx, off));
    pminy = fminf(pminy, __shfl_xor(pminy, off));
    pmaxx = fmaxf(pmaxx, __shfl_xor(pmaxx, off));
    pmaxy = fmaxf(pmaxy, __shfl_xor(pmaxy, off));
    tminx = fminf(tminx, __shfl_xor(tminx, off));
    tminy = fminf(tminy, __shfl_xor(tminy, off));
    tmaxx = fmaxf(tmaxx, __shfl_xor(tmaxx, off));
    tmaxy = fmaxf(tmaxy, __shfl_xor(tmaxy, off));
  }
  if (lane == 0) {
    MMX[wave][0] = pminx; MMX[wave][1] = pminy; MMX[wave][2] = pmaxx; MMX[wave][3] = pmaxy;
    MMX[wave][4] = tminx; MMX[wave][5] = tminy; MMX[wave][6] = tmaxx; MMX[wave][7] = tmaxy;
  }
  __syncthreads();

  float* __restrict__ prow = partials + (long)(img * NCHUNK + chunk) * NQ;

  // --- matrix-unit block reduce of the 7 sums (V_WMMA_F32_16X16X4_F32, full f32) ---
  // A = ones(16x4).  B (4x16): lane = column n (mod 16), K along {halfwave, vgpr}:
  //   B[k][n] with n = lane&15, k = (lane>>4)*2 + vgpr   (ISA 7.12.2 column striping).
  // D[m][n] += sum_k B[k][n]  ==> accumulated over all 64 groups: D[m][n] = sum_t P[t][n].
  // 4 independent accumulator chains so the WMMAs pipeline instead of RAW-stalling.
  if (wave == 0) {
    v8f d0 = {0,0,0,0,0,0,0,0}, d1 = d0, d2 = d0, d3 = d0;
    v2f a; a[0] = 1.0f; a[1] = 1.0f;
    const int n = lane & 15;
    const int kb = (lane >> 4) * 2;
    for (int g = 0; g < NTHREADS / 4; g += 4) {
      v2f b0, b1, b2, b3;
      b0[0] = P[(g + 0) * 4 + kb + 0][n]; b0[1] = P[(g + 0) * 4 + kb + 1][n];
      b1[0] = P[(g + 1) * 4 + kb + 0][n]; b1[1] = P[(g + 1) * 4 + kb + 1][n];
      b2[0] = P[(g + 2) * 4 + kb + 0][n]; b2[1] = P[(g + 2) * 4 + kb + 1][n];
      b3[0] = P[(g + 3) * 4 + kb + 0][n]; b3[1] = P[(g + 3) * 4 + kb + 1][n];
      d0 = __builtin_amdgcn_wmma_f32_16x16x4_f32(false, a, false, b0, (short)0, d0, false, false);
      d1 = __builtin_amdgcn_wmma_f32_16x16x4_f32(false, a, false, b1, (short)0, d1, false, false);
      d2 = __builtin_amdgcn_wmma_f32_16x16x4_f32(false, a, false, b2, (short)0, d2, false, false);
      d3 = __builtin_amdgcn_wmma_f32_16x16x4_f32(false, a, false, b3, (short)0, d3, false, false);
    }
    const float tot = (d0[0] + d1[0]) + (d2[0] + d3[0]);  // D[M=0][N=lane]
    if (lane < 8) prow[lane] = tot;                        // block total of quantity `lane`
  }

  // --- cross-wave min/max combine (thread 0) ---
  if (tid == 0) {
    float mm[8];
#pragma unroll
    for (int j = 0; j < 8; ++j) mm[j] = MMX[0][j];
#pragma unroll
    for (int w2 = 1; w2 < 8; ++w2) {
      mm[0] = fminf(mm[0], MMX[w2][0]); mm[1] = fminf(mm[1], MMX[w2][1]);
      mm[2] = fmaxf(mm[2], MMX[w2][2]); mm[3] = fmaxf(mm[3], MMX[w2][3]);
      mm[4] = fminf(mm[4], MMX[w2][4]); mm[5] = fminf(mm[5], MMX[w2][5]);
      mm[6] = fmaxf(mm[6], MMX[w2][6]); mm[7] = fmaxf(mm[7], MMX[w2][7]);
    }
#pragma unroll
    for (int j = 0; j < 8; ++j) prow[8 + j] = mm[j];
  }
}

__global__ __launch_bounds__(64) void
stage2(const float* __restrict__ partials, float* __restrict__ perimg) {
  __shared__ float R[NCHUNK][NQ];
  const int t = threadIdx.x;
  const int img = blockIdx.x;
  const float* __restrict__ row = partials + (long)(img * NCHUNK + t) * NQ;
#pragma unroll
  for (int q = 0; q < NQ; ++q) R[t][q] = row[q];
  __syncthreads();
  for (int s = NCHUNK / 2; s > 0; s >>= 1) {
    if (t < s) {
#pragma unroll
      for (int q = 0; q < 7; ++q) R[t][q] += R[t + s][q];
      R[t][8]  = fminf(R[t][8],  R[t + s][8]);  R[t][9]  = fminf(R[t][9],  R[t + s][9]);
      R[t][10] = fmaxf(R[t][10], R[t + s][10]); R[t][11] = fmaxf(R[t][11], R[t + s][11]);
      R[t][12] = fminf(R[t][12], R[t + s][12]); R[t][13] = fminf(R[t][13], R[t + s][13]);
      R[t][14] = fmaxf(R[t][14], R[t + s][14]); R[t][15] = fmaxf(R[t][15], R[t + s][15]);
    }
    __syncthreads();
  }
  if (t == 0) {
    const float eps = 1e-8f;
    const float S_p = R[0][0], S_t = R[0][1], S_pt = R[0][2];
    const float S_xp = R[0][3], S_yp = R[0][4], S_xt = R[0][5], S_yt = R[0][6];
    const float m = fminf(S_p, S_t), M = fmaxf(S_p, S_t);
    float al = (3.0f * m + M) / (3.0f * M + m);
    al = fminf(fmaxf(al, 1.0f / 3.0f), 1.0f);
    const float iou = S_pt / (S_p + S_t - S_pt);
    const float pcx = S_xp / (S_p + eps), pcy = S_yp / (S_p + eps);
    const float tcx = S_xt / (S_t + eps), tcy = S_yt / (S_t + eps);
    const float d2 = (pcx - tcx) * (pcx - tcx) + (pcy - tcy) * (pcy - tcy);
    float pminx = R[0][8], pminy = R[0][9], pmaxx = R[0][10], pmaxy = R[0][11];
    float tminx = R[0][12], tminy = R[0][13], tmaxx = R[0][14], tmaxy = R[0][15];
    if (!(pminx < 1e30f)) { pminx = 0.f; pminy = 0.f; pmaxx = 1.f; pmaxy = 1.f; }
    if (!(tminx < 1e30f)) { tminx = 0.f; tminy = 0.f; tmaxx = 1.f; tmaxy = 1.f; }
    const float mnx = fminf(pminx, tminx), mny = fminf(pminy, tminy);
    const float mxx = fmaxf(pmaxx, tmaxx), mxy = fmaxf(pmaxy, tmaxy);
    const float c2 = (mxx - mnx) * (mxx - mnx) + (mxy - mny) * (mxy - mny) + eps;
    const float A = S_t / (float)HWPX;
    const float gamma = sqrtf(0.01f / (A + eps));   // pow(.., 0.5)
    const float Dl = d2 / c2 * gamma;
    perimg[img * 3 + 0] = al * iou;
    perimg[img * 3 + 1] = Dl;
    perimg[img * 3 + 2] = iou;
  }
}

__global__ __launch_bounds__(32) void
stage3(const float* __restrict__ perimg, const int* __restrict__ warm,
       const int* __restrict__ epoch, float* __restrict__ out) {
  const int t = threadIdx.x;
  float l3 = perimg[t * 3 + 0];
  float Dl = perimg[t * 3 + 1];
  float io = perimg[t * 3 + 2];
#pragma unroll
  for (int off = 16; off > 0; off >>= 1) {
    l3 += __shfl_xor(l3, off);
    Dl += __shfl_xor(Dl, off);
    io += __shfl_xor(io, off);
  }
  if (t == 0) {
    const float invB = 1.0f / (float)BB;
    float loss;
    if (*epoch > *warm) loss = 1.0f - l3 * invB + 0.5f * (Dl * invB);
    else                loss = 1.0f - io * invB;
    out[0] = loss;
  }
}

extern "C" void kernel_launch(void* const* d_in, const int* in_sizes, int n_in,
                              void* d_out, int out_size, void* d_ws, size_t ws_size,
                              hipStream_t stream) {
  const float* pred_log = (const float*)d_in[0];
  const float* target   = (const float*)d_in[1];
  const int* warm  = (const int*)d_in[2];
  const int* epoch = (const int*)d_in[3];

  float* partials = (float*)d_ws;                        // BB*NCHUNK*NQ = 32768 f32
  float* perimg   = partials + (size_t)BB * NCHUNK * NQ; // 32*3 f32

  stage1<<<dim3(NCHUNK, BB), NTHREADS, 0, stream>>>(pred_log, target, partials);
  stage2<<<dim3(BB), NCHUNK, 0, stream>>>(partials, perimg);
  stage3<<<dim3(1), 32, 0, stream>>>(perimg, warm, epoch, (float*)d_out);
}